// StreamingDurationProjector_70506183131191
// MI455X (gfx1250) — compile-verified
//
#include <hip/hip_runtime.h>
#include <hip/hip_bf16.h>

#define B_DIM   2048
#define T_DIM   4096
#define ROWS    32                 // rows per block == lanes per wave32
#define CHUNK   32                 // timesteps per pipeline stage
#define NCHUNK  (T_DIM / CHUNK)    // 128
#define STRIDE  36                 // floats per staged row (32 + 4 TDM pad DWORDs)
#define ARR     (ROWS * STRIDE)    // 1152 floats per staged array
#define QPT     8                  // float4 quads per row per chunk (CHUNK/4)
#define BUFBYTES (6 * ARR * 4)     // 27648 bytes per input buffer

#define BUDGET_POS 24.0f
#define BUDGET_NEG 24.0f
#define DECAY      0.25f
#define THRESH     0.5f

typedef uint32_t u32x4 __attribute__((ext_vector_type(4)));
typedef uint32_t u32x8 __attribute__((ext_vector_type(8)));

__device__ __forceinline__ uint32_t lds_lo32(const void* p) {
    // generic LDS pointer: addr[31:0] is the hardware LDS byte offset (ISA 10.2)
    return (uint32_t)(uintptr_t)p;
}

// TDM: DMA one 2D tile (tile_dim0 x tile_dim1) global -> LDS. Tracked by TENSORcnt.
__device__ __forceinline__ void tdm_load(u32x4 g0, u32x8 g1) {
    asm volatile("tensor_load_to_lds %0, %1"
                 :: "s"(g0), "s"(g1)
                 : "memory");
}

// D# group 0: count=1 | lds_addr | global_addr(57b) | type=2
__device__ __forceinline__ u32x4 make_g0(uint32_t lds_word, uint64_t gaddr) {
    u32x4 g0;
    g0.x = 1u;                                           // count=1, user mode, no gather
    g0.y = lds_word;                                     // LDS byte address
    g0.z = (uint32_t)gaddr;                              // global_addr[31:0]
    g0.w = ((uint32_t)(gaddr >> 32) & 0x01FFFFFFu)       // global_addr[56:32]
           | 0x80000000u;                                // type=2 ("image") in bits 127:126
    return g0;
}

// One scan step of the reference recurrence (algebraically minimized, exact).
__device__ __forceinline__ void scan_step(float ex, float sc, float sp, float co,
                                          float bd, float pf,
                                          float& carry, float& prefix,
                                          float& proj, float& hit) {
    // (sp>0.5)||(co>0.5)  ==  max(sp,co)>0.5   (exact: '>' monotone under max)
    const bool  active = fmaxf(sp, co) > 0.5f;
    const float anchor = fmaxf(sc, 1.0f);
    const float total  = fmaxf(ex + carry, 0.0f);
    const float rnd    = floorf(total + 0.5f);
    const float lower  = fmaxf(ceilf(anchor - (BUDGET_NEG + prefix)), 1.0f);
    const float upper  = fmaxf(floorf(anchor + (BUDGET_POS - prefix)), lower);
    // lower >= 1 already, so reference's max(rnd,1) is subsumed by max(rnd,lower)
    const float frames = fminf(fmaxf(rnd, lower), upper);

    proj = active ? frames : anchor;
    float nprefix = active ? (prefix + (frames - anchor)) : prefix;
    float ncarry  = active ? (total - frames) : carry;

    // keep >= vs > distinction exact; no max-merge here
    const bool event = active && ((bd >= THRESH) || (pf > 0.5f));
    hit = event ? 1.0f : 0.0f;

    carry  = event ? (ncarry * DECAY) : ncarry;
    prefix = event ? fminf(fmaxf(nprefix * DECAY, -BUDGET_NEG), BUDGET_POS) : nprefix;
}

extern "C" __global__ __launch_bounds__(ROWS)
void StreamingDurationProjector_kernel(
        const float* __restrict__ ex_p,  const float* __restrict__ src_p,
        const float* __restrict__ sp_p,  const float* __restrict__ co_p,
        const float* __restrict__ bd_p,  const float* __restrict__ pf_p,
        const float* __restrict__ carry_init, const float* __restrict__ prefix_init,
        float* __restrict__ proj_o, float* __restrict__ hit_o,
        float* __restrict__ dec_o,  float* __restrict__ carry_o,
        float* __restrict__ prefix_o)
{
    // LDS (floats): buf0 in [6*ARR] | buf1 in [6*ARR] | proj stage [ARR] | hit stage [ARR]
    extern __shared__ float smem[];

    const int tid  = threadIdx.x;          // 0..31, one lane per row
    const int row0 = blockIdx.x * ROWS;

    const uint32_t lds_base = lds_lo32(&smem[0]);

    float carry  = carry_init[row0 + tid];
    float prefix = prefix_init[row0 + tid];

    // ---- constant D# group 1 (shared by all 6 arrays, all chunks) ----
    // data_size=2 (4B), pad_enable, pad_interval=4 (every 32 DWORDs), pad_amount=3 (4 DWORDs)
    u32x8 g1;
    g1.s0 = (2u << 16) | (1u << 20) | (4u << 22) | (3u << 25);
    g1.s1 = 0x7FFFFFFFu << 16;                             // tensor_dim0[15:0]
    g1.s2 = (0x7FFFFFFFu >> 16) | (0x7FFFFFFFu << 16);     // dim0[31:16] | dim1[15:0]
    g1.s3 = (0x7FFFFFFFu >> 16) | ((uint32_t)CHUNK << 16); // dim1[31:16] | tile_dim0=32
    g1.s4 = (uint32_t)ROWS;                                // tile_dim1=32, tile_dim2=0
    g1.s5 = (uint32_t)T_DIM;                               // tensor_dim0_stride = 4096
    g1.s6 = 0u;
    g1.s7 = 0u;

    // Per-array global cursor for the NEXT chunk to load (wave-uniform -> SALU)
    uint64_t ga[6];
    ga[0] = (uint64_t)(uintptr_t)(ex_p  + (size_t)row0 * T_DIM);
    ga[1] = (uint64_t)(uintptr_t)(src_p + (size_t)row0 * T_DIM);
    ga[2] = (uint64_t)(uintptr_t)(sp_p  + (size_t)row0 * T_DIM);
    ga[3] = (uint64_t)(uintptr_t)(co_p  + (size_t)row0 * T_DIM);
    ga[4] = (uint64_t)(uintptr_t)(bd_p  + (size_t)row0 * T_DIM);
    ga[5] = (uint64_t)(uintptr_t)(pf_p  + (size_t)row0 * T_DIM);

    // Coalesced writeback map: quad q = k*32 + tid ; r = 4k + tid/8 ; tq = tid%8
    const int rq = tid >> 3;
    const int tq = tid & 7;

    float* sproj = smem + 12 * ARR;
    float* shit  = smem + 13 * ARR;

    // ---- prologue: TDM chunk 0 into buf0 ----
    #pragma unroll
    for (int a = 0; a < 6; ++a) {
        tdm_load(make_g0(lds_base + (uint32_t)(a * ARR * 4), ga[a]), g1);
        ga[a] += CHUNK * 4;    // advance cursor 128B
    }

    for (int c = 0; c < NCHUNK; ++c) {
        const int t0  = c * CHUNK;
        const int cur = c & 1;

        if (c + 1 < NCHUNK) {
            // prior compute's LDS reads must retire before DMA reuses that buffer
            asm volatile("s_wait_dscnt 0x0" ::: "memory");
            const uint32_t bbase = lds_base + (uint32_t)(((c + 1) & 1) * BUFBYTES);
            #pragma unroll
            for (int a = 0; a < 6; ++a) {
                tdm_load(make_g0(bbase + (uint32_t)(a * ARR * 4), ga[a]), g1);
                ga[a] += CHUNK * 4;
            }
            // TDM ops retire in order: <=6 outstanding => chunk c fully in LDS,
            // chunk c+1 still streaming behind the compute below.
            __builtin_amdgcn_s_wait_tensorcnt(6);
        } else {
            __builtin_amdgcn_s_wait_tensorcnt(0);
        }

        // ---- serial scan: lane = row, float4 LDS traffic (ds_load_b128) ----
        const float* b = smem + cur * 6 * ARR + tid * STRIDE;
        #pragma unroll
        for (int qq = 0; qq < QPT; ++qq) {
            const float4 vex = *(const float4*)&b[0 * ARR + 4 * qq];
            const float4 vsc = *(const float4*)&b[1 * ARR + 4 * qq];
            const float4 vsp = *(const float4*)&b[2 * ARR + 4 * qq];
            const float4 vco = *(const float4*)&b[3 * ARR + 4 * qq];
            const float4 vbd = *(const float4*)&b[4 * ARR + 4 * qq];
            const float4 vpf = *(const float4*)&b[5 * ARR + 4 * qq];

            float4 vproj, vhit;
            scan_step(vex.x, vsc.x, vsp.x, vco.x, vbd.x, vpf.x, carry, prefix, vproj.x, vhit.x);
            scan_step(vex.y, vsc.y, vsp.y, vco.y, vbd.y, vpf.y, carry, prefix, vproj.y, vhit.y);
            scan_step(vex.z, vsc.z, vsp.z, vco.z, vbd.z, vpf.z, carry, prefix, vproj.z, vhit.z);
            scan_step(vex.w, vsc.w, vsp.w, vco.w, vbd.w, vpf.w, carry, prefix, vproj.w, vhit.w);

            *(float4*)&sproj[tid * STRIDE + 4 * qq] = vproj;
            *(float4*)&shit[tid * STRIDE + 4 * qq]  = vhit;
        }

        __syncthreads();   // single-wave WG: cheap fence ordering DS stores vs cross-lane reads

        // ---- coalesced float4 writeback of proj / hit / dec(==hit) ----
        #pragma unroll
        for (int k = 0; k < QPT; ++k) {
            const int r = 4 * k + rq;
            const size_t go = (size_t)(row0 + r) * T_DIM + t0 + 4 * tq;
            const float4 vp = *(const float4*)&sproj[r * STRIDE + 4 * tq];
            const float4 vh = *(const float4*)&shit[r * STRIDE + 4 * tq];
            *(float4*)&proj_o[go] = vp;
            *(float4*)&hit_o[go]  = vh;
            *(float4*)&dec_o[go]  = vh;
        }
    }

    carry_o[row0 + tid]  = carry;
    prefix_o[row0 + tid] = prefix;
}

extern "C" void kernel_launch(void* const* d_in, const int* in_sizes, int n_in,
                              void* d_out, int out_size, void* d_ws, size_t ws_size,
                              hipStream_t stream) {
    (void)in_sizes; (void)n_in; (void)out_size; (void)d_ws; (void)ws_size;

    const float* ex_p  = (const float*)d_in[0];
    const float* src_p = (const float*)d_in[1];
    const float* sp_p  = (const float*)d_in[2];
    const float* co_p  = (const float*)d_in[3];
    const float* bd_p  = (const float*)d_in[4];
    const float* pf_p  = (const float*)d_in[5];
    const float* c0_p  = (const float*)d_in[6];
    const float* p0_p  = (const float*)d_in[7];

    float* out      = (float*)d_out;
    float* proj_o   = out;
    float* hit_o    = proj_o + (size_t)B_DIM * T_DIM;
    float* dec_o    = hit_o  + (size_t)B_DIM * T_DIM;
    float* carry_o  = dec_o  + (size_t)B_DIM * T_DIM;
    float* prefix_o = carry_o + B_DIM;

    const dim3 grid(B_DIM / ROWS);   // 64 blocks -> 64 WGPs busy
    const dim3 block(ROWS);          // 32 threads = 1 wave32
    const size_t shmem = (size_t)14 * ARR * sizeof(float);  // 64,512 B (< 320 KB/WGP)

    StreamingDurationProjector_kernel<<<grid, block, shmem, stream>>>(
        ex_p, src_p, sp_p, co_p, bd_p, pf_p, c0_p, p0_p,
        proj_o, hit_o, dec_o, carry_o, prefix_o);
}